// GraphConvolution_1580547969797
// MI455X (gfx1250) — compile-verified
//
#include <hip/hip_runtime.h>

typedef float v2f  __attribute__((ext_vector_type(2)));
typedef float v8f  __attribute__((ext_vector_type(8)));
typedef unsigned int u32x4 __attribute__((ext_vector_type(4)));
typedef int   i32x4 __attribute__((ext_vector_type(4)));
typedef int   i32x8 __attribute__((ext_vector_type(8)));

#define D       128
#define TILE_M  16
#define LDS_PAD 4   // 132-float row stride -> lanes hit banks 0,4,...,60 (conflict-free)

// ---------------------------------------------------------------------------
// Phase 1: support = input @ W  via V_WMMA_F32_16X16X4_F32
// block = 256 threads (8 wave32); one block computes a 16x128 slab of support.
// A tile (16x128 f32) is DMA'd into LDS by the Tensor Data Mover with
// pad_interval=128 DW / pad_amount=4 DW, reproducing the 132-float padded rows.
// ---------------------------------------------------------------------------
__global__ __launch_bounds__(256)
void gcn_gemm_wmma(const float* __restrict__ A, const float* __restrict__ W,
                   float* __restrict__ S, int M) {
  __shared__ float As[TILE_M][D + LDS_PAD];

  const int tid = threadIdx.x;
  const int m0  = blockIdx.x * TILE_M;

  if (tid < 32) {
    // Wave 0 programs the TDM: 2D tile load Global -> LDS.
    const unsigned lds_base = (unsigned)(size_t)(&As[0][0]); // flat->LDS: addr[31:0]
    const unsigned long long ga =
        (unsigned long long)(size_t)A + (unsigned long long)m0 * D * 4ull;
    const unsigned rem = (unsigned)(M - m0);  // rows available from tile start

    // D# group 0: count=1 | lds_addr | global_addr[56:0] | type=2
    u32x4 g0;
    g0[0] = 1u;
    g0[1] = lds_base;
    g0[2] = (unsigned)(ga & 0xffffffffu);
    g0[3] = (unsigned)((ga >> 32) & 0x1ffffffu) | (2u << 30);

    // D# group 1:
    //  w0: wg_mask=0 | data_size=2 (4B) | pad_enable | pad_interval=6 (128 DW)
    //      | pad_amount=3 (4 DW)
    //  tensor_dim0=128, tensor_dim1=rem (OOB rows read as zero),
    //  tile_dim0=128, tile_dim1=16, tensor_dim0_stride=128
    i32x8 g1;
    g1[0] = (int)((2u << 16) | (1u << 20) | (6u << 22) | (3u << 25));
    g1[1] = (int)(128u << 16);                   // tensor_dim0[15:0]
    g1[2] = (int)((rem & 0xffffu) << 16);        // tensor_dim0[31:16]=0 | tensor_dim1[15:0]
    g1[3] = (int)((rem >> 16) | (128u << 16));   // tensor_dim1[31:16] | tile_dim0=128
    g1[4] = (int)TILE_M;                         // tile_dim1=16 | tile_dim2=0
    g1[5] = 128;                                 // tensor_dim0_stride[31:0]
    g1[6] = 0;                                   // stride[47:32] | dim1_stride lo
    g1[7] = 0;

    const i32x4 gz4 = {0, 0, 0, 0};              // groups 2/3 unused (2D tensor)
    const i32x8 gz8 = {0, 0, 0, 0, 0, 0, 0, 0};  // trailing group (6-arg form)
    __builtin_amdgcn_tensor_load_to_lds(g0, g1, gz4, gz4, gz8, 0);
    __builtin_amdgcn_s_wait_tensorcnt(0);        // TDM writes to LDS complete
  }
  __syncthreads();                               // publish tile to all 8 waves

  const int wave = tid >> 5;              // 8 waves -> 8 N-tiles of 16 cols
  const int lane = tid & 31;
  const int n0   = wave * 16;
  const int mrow = lane & 15;             // A fragment: M index for this lane
  const int koff = (lane >> 4) << 1;      // 0 (lanes 0-15) or 2 (lanes 16-31)
  const int ncol = n0 + (lane & 15);      // B/C/D fragment: N index for this lane

  v8f acc = {};
  #pragma unroll
  for (int k = 0; k < D; k += 4) {
    v2f a, b;
    // A 16x4 fragment (ISA 7.12.2): VGPR0=K(base), VGPR1=K(base+1)
    a.x = As[mrow][k + koff];
    a.y = As[mrow][k + koff + 1];
    // B 4x16 fragment: row K striped across lanes by N (coalesced 64B/half-wave)
    b.x = W[(size_t)(k + koff)     * D + ncol];
    b.y = W[(size_t)(k + koff + 1) * D + ncol];
    acc = __builtin_amdgcn_wmma_f32_16x16x4_f32(
        /*neg_a=*/false, a, /*neg_b=*/false, b,
        /*c_mod=*/(short)0, acc, /*reuse_a=*/false, /*reuse_b=*/false);
  }

  // D fragment: VGPR j -> M = j (lanes 0-15) or 8+j (lanes 16-31)
  const int mhalf = (lane >> 4) << 3;
  #pragma unroll
  for (int j = 0; j < 8; ++j) {
    const int mr = m0 + mhalf + j;
    if (mr < M) S[(size_t)mr * D + ncol] = acc[j];
  }
}

// ---------------------------------------------------------------------------
// Phase 2a: out[n, d] = bias[d]   (reference adds bias AFTER segment_sum)
// float4-vectorized: D/4 = 32 float4 per row.
// ---------------------------------------------------------------------------
__global__ __launch_bounds__(256)
void gcn_init_bias(const float4* __restrict__ bias4, float4* __restrict__ out4,
                   size_t total4) {
  const size_t i = (size_t)blockIdx.x * 256 + threadIdx.x;
  if (i < total4) out4[i] = bias4[i & 31];
}

// ---------------------------------------------------------------------------
// Phase 2b: wave-per-edge scatter-add.
// Lane l owns features [4l, 4l+4): one coalesced float4 gather from L2-resident
// support, then 4 hardware fp32 atomics into out[rows[e]].
// ---------------------------------------------------------------------------
__global__ __launch_bounds__(256)
void gcn_scatter(const float* __restrict__ S, const float* __restrict__ vals,
                 const int* __restrict__ rows, const int* __restrict__ cols,
                 float* __restrict__ out, int E) {
  const long long gid = (long long)blockIdx.x * 256 + threadIdx.x;
  const int e    = (int)(gid >> 5);
  const int lane = (int)(gid & 31);
  if (e >= E) return;

  const int   r = rows[e];
  const int   c = cols[e];
  const float v = vals[e];

  const float4 s = *reinterpret_cast<const float4*>(&S[(size_t)c * D + lane * 4]);
  float* dst = &out[(size_t)r * D + lane * 4];
  unsafeAtomicAdd(dst + 0, v * s.x);
  unsafeAtomicAdd(dst + 1, v * s.y);
  unsafeAtomicAdd(dst + 2, v * s.z);
  unsafeAtomicAdd(dst + 3, v * s.w);
}

// ---------------------------------------------------------------------------
extern "C" void kernel_launch(void* const* d_in, const int* in_sizes, int n_in,
                              void* d_out, int out_size, void* d_ws, size_t ws_size,
                              hipStream_t stream) {
  const float* input     = (const float*)d_in[0];  // [N, 128] f32
  const float* weight    = (const float*)d_in[1];  // [128, 128] f32
  const float* bias      = (const float*)d_in[2];  // [128] f32
  const float* edge_vals = (const float*)d_in[3];  // [E] f32
  const int*   rows      = (const int*)d_in[4];    // [E] i32
  const int*   cols      = (const int*)d_in[5];    // [E] i32
  float*       out       = (float*)d_out;          // [N, 128] f32
  float*       support   = (float*)d_ws;           // scratch: N*128 f32 (25.6 MB)

  const int M = in_sizes[0] / D;   // 50000 nodes
  const int E = in_sizes[3];       // 800000 edges

  // 1) support = input @ W  (WMMA f32, TDM-staged A tiles)
  gcn_gemm_wmma<<<(M + TILE_M - 1) / TILE_M, 256, 0, stream>>>(input, weight,
                                                               support, M);

  // 2a) out = broadcast(bias)
  const size_t total4 = (size_t)M * (D / 4);
  gcn_init_bias<<<(unsigned)((total4 + 255) / 256), 256, 0, stream>>>(
      (const float4*)bias, (float4*)out, total4);

  // 2b) out[rows[e]] += edge_vals[e] * support[cols[e]]
  const long long threads = (long long)E * 32;
  gcn_scatter<<<(unsigned)((threads + 255) / 256), 256, 0, stream>>>(
      support, edge_vals, rows, cols, out, E);
}